// LG_Encoder_15238543966835
// MI455X (gfx1250) — compile-verified
//
#include <hip/hip_runtime.h>
#include <math.h>

#define F 128
#define NNODES 100000
#define NEDGES 200000
#define NLG 600000
#define NGRAPHS 4096
#define NITER 3

typedef __attribute__((ext_vector_type(2))) float v2f;
typedef __attribute__((ext_vector_type(8))) float v8f;

// ---------------------------------------------------------------------------
// fp32 WMMA GEMM: C[M,N] = act(alpha * A@B(^T) + bias) (+= if accum)
// One 16x64 C macro-tile per wave32 (4 accumulators share one A fragment),
// K stepped by 4 via V_WMMA_F32_16X16X4_F32, software-pipelined loads.
// Fragment layout per CDNA5 ISA 7.12.2:
//   A 16x4 : lane%16 = M row; VGPR0/1 = K {0,1} (lanes 0-15) / {2,3} (16-31)
//   B 4x16 : lane%16 = N col; VGPR0/1 = K {0,1} (lanes 0-15) / {2,3} (16-31)
//   D 16x16: VGPR r -> row r (lanes 0-15) / r+8 (lanes 16-31), col = lane%16
// Requires: M%16==0, N%64==0, K%4==0 (all shapes here satisfy this).
// ---------------------------------------------------------------------------
template <int TRANSB>
__global__ void k_gemm_t(const float* __restrict__ A, const float* __restrict__ B,
                         float* __restrict__ C, int M, int N, int K,
                         const float* __restrict__ bias, float alpha,
                         int act, int accum) {
  int wid  = blockIdx.x * (blockDim.x >> 5) + (threadIdx.x >> 5);
  int lane = threadIdx.x & 31;
  int tn_cnt = N >> 6;                       // 64-wide N groups
  int tiles  = (M >> 4) * tn_cnt;
  if (wid >= tiles) return;
  int tm = wid / tn_cnt;
  int tn = wid - tm * tn_cnt;
  int m0 = tm << 4, n0 = tn << 6;
  int lh = lane & 15;
  int hi = lane >> 4;

  const v8f vzero = {0.f, 0.f, 0.f, 0.f, 0.f, 0.f, 0.f, 0.f};
  v8f accs[4] = {vzero, vzero, vzero, vzero};

  const float* ap = A + (size_t)(m0 + lh) * K + (hi << 1);
  const float* bp[4];
  size_t bstep;
  if (TRANSB) {
    bp[0] = B + (size_t)(n0 + lh) * K + (hi << 1);
    bp[1] = bp[0] + (size_t)16 * K;
    bp[2] = bp[0] + (size_t)32 * K;
    bp[3] = bp[0] + (size_t)48 * K;
    bstep = 4;
  } else {
    bp[0] = B + (size_t)(hi << 1) * N + n0 + lh;
    bp[1] = bp[0] + 16;
    bp[2] = bp[0] + 32;
    bp[3] = bp[0] + 48;
    bstep = (size_t)4 * N;
  }

  auto ldA = [&](const float* p) {
    float2 t = *(const float2*)p; v2f r; r.x = t.x; r.y = t.y; return r;
  };
  auto ldB = [&](const float* p) {
    v2f r;
    if (TRANSB) { float2 t = *(const float2*)p; r.x = t.x; r.y = t.y; }
    else        { r.x = p[0]; r.y = p[N]; }
    return r;
  };

  // software pipeline: fragments for step k loaded in step k-1
  v2f a = ldA(ap);
  v2f b[4] = {ldB(bp[0]), ldB(bp[1]), ldB(bp[2]), ldB(bp[3])};
  for (int k0 = 4; k0 < K; k0 += 4) {
    ap += 4;
#pragma unroll
    for (int s = 0; s < 4; ++s) bp[s] += bstep;
    v2f an = ldA(ap);
    v2f bn[4] = {ldB(bp[0]), ldB(bp[1]), ldB(bp[2]), ldB(bp[3])};
#pragma unroll
    for (int s = 0; s < 4; ++s)
      accs[s] = __builtin_amdgcn_wmma_f32_16x16x4_f32(false, a, false, b[s],
                                                      (short)0, accs[s], false, false);
    a = an;
#pragma unroll
    for (int s = 0; s < 4; ++s) b[s] = bn[s];
  }
#pragma unroll
  for (int s = 0; s < 4; ++s)
    accs[s] = __builtin_amdgcn_wmma_f32_16x16x4_f32(false, a, false, b[s],
                                                    (short)0, accs[s], false, false);

#pragma unroll
  for (int s = 0; s < 4; ++s) {
    int col = n0 + (s << 4) + lh;
    float bv = bias ? bias[col] : 0.f;
#pragma unroll
    for (int r = 0; r < 8; ++r) {
      int row = m0 + r + (hi << 3);
      float v = accs[s][r] * alpha + bv;
      if (act == 1)      v = tanhf(v);
      else if (act == 2) v = v > 0.f ? v : (expf(v) - 1.f);
      size_t idx = (size_t)row * N + col;
      if (accum) C[idx] += v; else C[idx] = v;
    }
  }
}

// ------------------------- small utility kernels ---------------------------
__global__ void k_zero(float* p, size_t n) {
  size_t i = (size_t)blockIdx.x * blockDim.x + threadIdx.x;
  if (i < n) p[i] = 0.f;
}

// ea = (edge_u[src] + edge_v[dst] + edge_attr @ lin_edge) / 3  (lin_edge in LDS)
__global__ void k_ea(const float* __restrict__ eu, const float* __restrict__ ev,
                     const float* __restrict__ eattr, const float* __restrict__ lin_edge,
                     const int* __restrict__ src, const int* __restrict__ dst,
                     float* __restrict__ ea, int E) {
  __shared__ float W[16 * F];
  for (int i = threadIdx.x; i < 16 * F; i += blockDim.x) W[i] = lin_edge[i];
  __syncthreads();
  int f  = threadIdx.x;                 // blockDim = 128
  int e0 = blockIdx.x * 16;
  int e1 = min(e0 + 16, E);
  for (int e = e0; e < e1; ++e) {
    int s = src[e], d = dst[e];
    float acc = eu[(size_t)s * F + f] + ev[(size_t)d * F + f];
    const float* ar = eattr + (size_t)e * 16;
#pragma unroll
    for (int k = 0; k < 16; ++k) acc += ar[k] * W[k * F + f];
    ea[(size_t)e * F + f] = acc * (1.f / 3.f);
  }
}

// nea = ea + sign(ea) * (noise / (||noise||+1e-12)) * 0.2 ; one wave per edge
__global__ void k_noise(const float* __restrict__ ea, const float* __restrict__ noise,
                        float* __restrict__ nea, int E) {
  int wid  = (blockIdx.x * blockDim.x + threadIdx.x) >> 5;
  int lane = threadIdx.x & 31;
  if (wid >= E) return;
  const float* nr = noise + (size_t)wid * F;
  float s = 0.f;
  for (int f = lane; f < F; f += 32) { float v = nr[f]; s += v * v; }
  for (int o = 16; o > 0; o >>= 1) s += __shfl_down(s, o, 32);
  s = __shfl(s, 0, 32);
  float inv = 0.2f / (sqrtf(s) + 1e-12f);
  const float* er = ea + (size_t)wid * F;
  float* orow = nea + (size_t)wid * F;
  for (int f = lane; f < F; f += 32) {
    float v  = er[f];
    float sg = (v > 0.f) ? 1.f : (v < 0.f ? -1.f : 0.f);
    orow[f] = v + sg * nr[f] * inv;
  }
}

__global__ void k_scatter1(const int* __restrict__ idx, float* __restrict__ acc, int n) {
  int e = blockIdx.x * blockDim.x + threadIdx.x;
  if (e < n) atomicAdd(&acc[idx[e]], 1.f);
}

__global__ void k_dinv_fin(float* d, int n) {
  int i = blockIdx.x * blockDim.x + threadIdx.x;
  if (i < n) d[i] = rsqrtf(d[i] + 1.f);
}

// out = h * dinv^2 + b   (self-loop term + bias, also zero-initializes out)
__global__ void k_init_out(const float* __restrict__ h, const float* __restrict__ dinv,
                           const float* __restrict__ b, float* __restrict__ out, int n) {
  size_t idx = (size_t)blockIdx.x * blockDim.x + threadIdx.x;
  if (idx >= (size_t)n * F) return;
  int i = (int)(idx >> 7), f = (int)(idx & 127);
  float dv = dinv[i];
  out[idx] = h[idx] * dv * dv + b[f];
}

// out[dst] += h[src] * dinv[src]*dinv[dst] ; one float4 quad per thread
__global__ void k_msg(const float* __restrict__ h, const float* __restrict__ dinv,
                      const int* __restrict__ src, const int* __restrict__ dst,
                      float* __restrict__ out, int nlg) {
  size_t idx = (size_t)blockIdx.x * blockDim.x + threadIdx.x;
  if (idx >= (size_t)nlg * (F / 4)) return;
  int e = (int)(idx >> 5), q = (int)(idx & 31);
  int s = src[e], d = dst[e];
  float w = dinv[s] * dinv[d];
  float4 hv = *(const float4*)(h + (size_t)s * F + q * 4);
  float* op = out + (size_t)d * F + q * 4;
  atomicAdd(op + 0, hv.x * w);
  atomicAdd(op + 1, hv.y * w);
  atomicAdd(op + 2, hv.z * w);
  atomicAdd(op + 3, hv.w * w);
}

__global__ void k_relu_res(const float* __restrict__ base, float* __restrict__ out, size_t n) {
  size_t idx = (size_t)blockIdx.x * blockDim.x + threadIdx.x;
  if (idx < n) out[idx] = fmaxf(base[idx] + out[idx], 0.f);
}

// per-row dots with W_rel / W_root ; one wave32 per row
__global__ void k_dots(const float* __restrict__ X, const float* __restrict__ Wrel,
                       const float* __restrict__ Wroot, float* __restrict__ srel,
                       float* __restrict__ sroot, int n) {
  int wid  = (blockIdx.x * blockDim.x + threadIdx.x) >> 5;
  int lane = threadIdx.x & 31;
  if (wid >= n) return;
  const float* row = X + (size_t)wid * F;
  float a = 0.f, b = 0.f;
  for (int f = lane; f < F; f += 32) { float v = row[f]; a += v * Wrel[f]; b += v * Wroot[f]; }
  for (int o = 16; o > 0; o >>= 1) { a += __shfl_down(a, o, 32); b += __shfl_down(b, o, 32); }
  if (lane == 0) { srel[wid] = a; sroot[wid] = b; }
}

__global__ void k_agg(const float* __restrict__ srel, const int* __restrict__ src,
                      const int* __restrict__ dst, float* __restrict__ aggs, int nlg) {
  int e = blockIdx.x * blockDim.x + threadIdx.x;
  if (e < nlg) atomicAdd(&aggs[dst[e]], srel[src[e]]);
}

__global__ void k_xc(const float* __restrict__ aggs, const float* __restrict__ sroot,
                     const float* __restrict__ brel, float* __restrict__ xc, int n) {
  int i = blockIdx.x * blockDim.x + threadIdx.x;
  if (i < n) xc[i] = aggs[i] + brel[0] + sroot[i];
}

// per-graph start offsets by binary search over the sorted batch array
__global__ void k_bounds(const int* __restrict__ batch, int* __restrict__ offs, int E, int G) {
  int g = blockIdx.x * blockDim.x + threadIdx.x;
  if (g > G) return;
  if (g == G) { offs[g] = E; return; }
  int lo = 0, hi = E;
  while (lo < hi) { int mid = (lo + hi) >> 1; if (batch[mid] < g) lo = mid + 1; else hi = mid; }
  offs[g] = lo;
}

__global__ void k_segsoftmax(const float* __restrict__ xc, const int* __restrict__ offs,
                             float* __restrict__ score, int G) {
  int g = blockIdx.x;
  int s0 = offs[g], s1 = offs[g + 1];
  if (s0 >= s1) return;
  __shared__ float red[128];
  float m = -3.4e38f;
  for (int e = s0 + threadIdx.x; e < s1; e += 128) m = fmaxf(m, xc[e]);
  red[threadIdx.x] = m; __syncthreads();
  for (int st = 64; st > 0; st >>= 1) {
    if (threadIdx.x < st) red[threadIdx.x] = fmaxf(red[threadIdx.x], red[threadIdx.x + st]);
    __syncthreads();
  }
  m = red[0]; __syncthreads();
  float s = 0.f;
  for (int e = s0 + threadIdx.x; e < s1; e += 128) { float ex = expf(xc[e] - m); score[e] = ex; s += ex; }
  red[threadIdx.x] = s; __syncthreads();
  for (int st = 64; st > 0; st >>= 1) {
    if (threadIdx.x < st) red[threadIdx.x] += red[threadIdx.x + st];
    __syncthreads();
  }
  float inv = 1.f / (red[0] + 1e-16f);
  for (int e = s0 + threadIdx.x; e < s1; e += 128) score[e] *= inv;
}

// pooled[g][f] = sum_{e in graph g} buf[e][f] * (score ? score[e] : 1)
__global__ void k_pool(const float* __restrict__ buf, const float* __restrict__ score,
                       const int* __restrict__ offs, float* __restrict__ pooled) {
  int g = blockIdx.x, f = threadIdx.x;
  int s0 = offs[g], s1 = offs[g + 1];
  float acc = 0.f;
  for (int e = s0; e < s1; ++e) acc += buf[(size_t)e * F + f] * (score ? score[e] : 1.f);
  pooled[(size_t)g * F + f] = acc;
}

// per-graph softmax over the 3 iteration scores
__global__ void k_iter_scores(const float* __restrict__ gout, const float* __restrict__ a,
                              const float* __restrict__ abias, float* __restrict__ itsc, int G) {
  int g = blockIdx.x * blockDim.x + threadIdx.x;
  if (g >= G) return;
  float sc[NITER];
  float mx = -3.4e38f;
  for (int t = 0; t < NITER; ++t) {
    const float* row = gout + ((size_t)t * G + g) * F;
    float d = 0.f;
    for (int f = 0; f < F; ++f) d += row[f] * a[f * NITER + t];
    sc[t] = d + abias[t];
    mx = fmaxf(mx, sc[t]);
  }
  float sum = 0.f;
  for (int t = 0; t < NITER; ++t) { sc[t] = expf(sc[t] - mx); sum += sc[t]; }
  for (int t = 0; t < NITER; ++t) itsc[(size_t)g * NITER + t] = sc[t] / sum;
}

// b0 = b0*w0 + b1*w1 + b2*w2  with per-edge graph weights
__global__ void k_combine(float* __restrict__ b0, const float* __restrict__ b1,
                          const float* __restrict__ b2, const int* __restrict__ batch,
                          const float* __restrict__ itsc, int n) {
  size_t idx = (size_t)blockIdx.x * blockDim.x + threadIdx.x;
  if (idx >= (size_t)n * F) return;
  int i = (int)(idx >> 7);
  const float* w = itsc + (size_t)batch[i] * NITER;
  b0[idx] = b0[idx] * w[0] + b1[idx] * w[1] + b2[idx] * w[2];
}

__global__ void k_norm_rows(float* __restrict__ z, int n, float eps) {
  int wid  = (blockIdx.x * blockDim.x + threadIdx.x) >> 5;
  int lane = threadIdx.x & 31;
  if (wid >= n) return;
  float* r = z + (size_t)wid * F;
  float s = 0.f;
  for (int f = lane; f < F; f += 32) { float v = r[f]; s += v * v; }
  for (int o = 16; o > 0; o >>= 1) s += __shfl_down(s, o, 32);
  s = __shfl(s, 0, 32);
  float inv = 1.f / (sqrtf(s) + eps);
  for (int f = lane; f < F; f += 32) r[f] *= inv;
}

__global__ void k_lse_row(const float* __restrict__ sim, float* __restrict__ term, int n) {
  int i = blockIdx.x;
  const float* row = sim + (size_t)i * n;
  __shared__ float red[256];
  float m = -3.4e38f;
  for (int j = threadIdx.x; j < n; j += 256) m = fmaxf(m, row[j]);
  red[threadIdx.x] = m; __syncthreads();
  for (int st = 128; st > 0; st >>= 1) {
    if (threadIdx.x < st) red[threadIdx.x] = fmaxf(red[threadIdx.x], red[threadIdx.x + st]);
    __syncthreads();
  }
  m = red[0]; __syncthreads();
  float s = 0.f;
  for (int j = threadIdx.x; j < n; j += 256) s += expf(row[j] - m);
  red[threadIdx.x] = s; __syncthreads();
  for (int st = 128; st > 0; st >>= 1) {
    if (threadIdx.x < st) red[threadIdx.x] += red[threadIdx.x + st];
    __syncthreads();
  }
  if (threadIdx.x == 0) term[i] = row[i] - (m + logf(red[0]));
}

__global__ void k_lse_col(const float* __restrict__ sim, float* __restrict__ term, int n) {
  int j = blockIdx.x * 32 + threadIdx.x;
  __shared__ float red[8][33];
  float m = -3.4e38f;
  for (int i = threadIdx.y; i < n; i += 8) m = fmaxf(m, sim[(size_t)i * n + j]);
  red[threadIdx.y][threadIdx.x] = m; __syncthreads();
  if (threadIdx.y == 0) {
    for (int r = 1; r < 8; ++r) m = fmaxf(m, red[r][threadIdx.x]);
    red[0][threadIdx.x] = m;
  }
  __syncthreads();
  m = red[0][threadIdx.x];
  __syncthreads();
  float s = 0.f;
  for (int i = threadIdx.y; i < n; i += 8) s += expf(sim[(size_t)i * n + j] - m);
  red[threadIdx.y][threadIdx.x] = s; __syncthreads();
  if (threadIdx.y == 0) {
    for (int r = 1; r < 8; ++r) s += red[r][threadIdx.x];
    term[j] = sim[(size_t)j * n + j] - (m + logf(s));
  }
}

__global__ void k_loss(const float* __restrict__ rt, const float* __restrict__ ct,
                       float* __restrict__ out, int n) {
  __shared__ float red[256];
  float s = 0.f;
  for (int i = threadIdx.x; i < n; i += 256) s += rt[i] + ct[i];
  red[threadIdx.x] = s; __syncthreads();
  for (int st = 128; st > 0; st >>= 1) {
    if (threadIdx.x < st) red[threadIdx.x] += red[threadIdx.x + st];
    __syncthreads();
  }
  if (threadIdx.x == 0) out[0] = -0.5f * red[0] / (float)n;
}

// ---------------------------------------------------------------------------
extern "C" void kernel_launch(void* const* d_in, const int* in_sizes, int n_in,
                              void* d_out_v, int out_size, void* d_ws, size_t ws_size,
                              hipStream_t stream) {
  if (n_in < 26) return;
  auto fp = [&](int i) { return (const float*)d_in[i]; };
  auto ip = [&](int i) { return (const int*)d_in[i]; };

  const float *cW1, *cW2, *cb1, *cb2, *lg_a, *lg_abias, *att_Wrel, *att_Wroot, *att_brel,
      *gcnW, *gcnb, *lingoutW, *lingoutb, *lin_edge, *lin_out, *lin_u, *lin_v,
      *top_Wrel, *top_Wroot, *top_brel, *x, *edge_attr, *noise;
  const int *edge_index, *lg_edge, *batch;

  if (in_sizes[2] == 2048) {           // dict insertion order flattening
    lin_u = fp(0); lin_v = fp(1); lin_edge = fp(2); gcnW = fp(3); gcnb = fp(4);
    att_Wrel = fp(5); att_brel = fp(6); att_Wroot = fp(7);
    lingoutW = fp(8); lingoutb = fp(9); lg_a = fp(10); lg_abias = fp(11);
    top_Wrel = fp(12); top_brel = fp(13); top_Wroot = fp(14);
    cW1 = fp(15); cb1 = fp(16); cW2 = fp(17); cb2 = fp(18); lin_out = fp(19);
  } else {                              // jax pytree (sorted dict keys)
    cW1 = fp(0); cW2 = fp(1); cb1 = fp(2); cb2 = fp(3);
    lg_a = fp(4); lg_abias = fp(5); att_Wrel = fp(6); att_Wroot = fp(7); att_brel = fp(8);
    gcnW = fp(9); gcnb = fp(10); lingoutW = fp(11); lingoutb = fp(12);
    lin_edge = fp(13); lin_out = fp(14); lin_u = fp(15); lin_v = fp(16);
    top_Wrel = fp(17); top_Wroot = fp(18); top_brel = fp(19);
  }
  x = fp(20); edge_attr = fp(21); noise = fp(22);
  edge_index = ip(23); lg_edge = ip(24); batch = ip(25);

  const int* src    = edge_index;
  const int* dst    = edge_index + NEDGES;
  const int* lg_src = lg_edge;
  const int* lg_dst = lg_edge + NLG;
  float* out = (float*)d_out_v;

  // -------- workspace layout --------
  float* ws = (float*)d_ws;
  size_t off = 0;
  auto alloc = [&](size_t n) { float* p = ws + off; off += n; return p; };
  const size_t NODEF = (size_t)NNODES * F;
  const size_t EF    = (size_t)NEDGES * F;
  const size_t GFsz  = (size_t)NGRAPHS * F;
  float* edge_u = alloc(NODEF);
  float* edge_v = alloc(NODEF);
  float* ea     = alloc(EF);
  float* nea    = alloc(EF);
  float* bufs   = alloc(6 * EF);           // bufs + (v*3+t)*EF
  float* hbuf   = alloc(EF);
  float* dinv   = alloc(NEDGES);
  float* srel   = alloc(NEDGES);
  float* sroot  = alloc(NEDGES);
  float* aggs   = alloc(NEDGES);
  float* xc     = alloc(NEDGES);
  float* score  = alloc(NEDGES);
  float* praw   = alloc(GFsz);
  float* goutb  = alloc(6 * GFsz);         // goutb + (v*3+t)*GFsz
  float* itsc   = alloc((size_t)2 * NGRAPHS * NITER);
  float* g1     = alloc(GFsz);
  float* g2     = alloc(GFsz);
  float* hp     = alloc(GFsz);
  float* z1     = alloc(GFsz);
  float* z2     = alloc(GFsz);
  float* sim    = alloc((size_t)NGRAPHS * NGRAPHS);
  float* rowt   = alloc(NGRAPHS);
  float* colt   = alloc(NGRAPHS);
  int*   offs   = (int*)(ws + off); off += NGRAPHS + 1;
  if (off * sizeof(float) > ws_size) return;   // workspace too small: bail

  auto gemm = [&](const float* A, const float* B, float* C, int M, int N, int K,
                  const float* bias, float alpha, int act, int accum, int transB) {
    int tiles  = (M / 16) * (N / 64);
    int blocks = (tiles + 7) / 8;
    if (transB)
      k_gemm_t<1><<<blocks, 256, 0, stream>>>(A, B, C, M, N, K, bias, alpha, act, accum);
    else
      k_gemm_t<0><<<blocks, 256, 0, stream>>>(A, B, C, M, N, K, bias, alpha, act, accum);
  };
  auto att_pool = [&](const float* buf, const float* Wrel, const float* Wroot,
                      const float* brel, float* pooled) {
    k_dots<<<(NEDGES + 7) / 8, 256, 0, stream>>>(buf, Wrel, Wroot, srel, sroot, NEDGES);
    k_zero<<<(NEDGES + 255) / 256, 256, 0, stream>>>(aggs, NEDGES);
    k_agg<<<(NLG + 255) / 256, 256, 0, stream>>>(srel, lg_src, lg_dst, aggs, NLG);
    k_xc<<<(NEDGES + 255) / 256, 256, 0, stream>>>(aggs, sroot, brel, xc, NEDGES);
    k_segsoftmax<<<NGRAPHS, 128, 0, stream>>>(xc, offs, score, NGRAPHS);
    k_pool<<<NGRAPHS, 128, 0, stream>>>(buf, score, offs, pooled);
  };

  // -------- phase A: edge embedding + noise view --------
  gemm(x, lin_u, edge_u, NNODES, F, F, nullptr, 1.f, 0, 0, 0);
  gemm(x, lin_v, edge_v, NNODES, F, F, nullptr, 1.f, 0, 0, 0);
  k_ea<<<(NEDGES + 15) / 16, 128, 0, stream>>>(edge_u, edge_v, edge_attr, lin_edge, src, dst, ea, NEDGES);
  k_noise<<<(NEDGES + 7) / 8, 256, 0, stream>>>(ea, noise, nea, NEDGES);

  // line-graph degree -> dinv  (same for both views / all iterations)
  k_zero<<<(NEDGES + 255) / 256, 256, 0, stream>>>(dinv, NEDGES);
  k_scatter1<<<(NLG + 255) / 256, 256, 0, stream>>>(lg_dst, dinv, NLG);
  k_dinv_fin<<<(NEDGES + 255) / 256, 256, 0, stream>>>(dinv, NEDGES);

  // per-graph edge ranges (batch is sorted)
  k_bounds<<<(NGRAPHS + 1 + 255) / 256, 256, 0, stream>>>(batch, offs, NEDGES, NGRAPHS);

  // -------- phase B: line-graph GCN, both views --------
  for (int v = 0; v < 2; ++v) {
    const float* base = (v == 0) ? ea : nea;
    const float* prev = base;
    for (int t = 0; t < NITER; ++t) {
      float* cur = bufs + ((size_t)(v * NITER + t)) * EF;
      gemm(prev, gcnW + (size_t)t * F * F, hbuf, NEDGES, F, F, nullptr, 1.f, 0, 0, 0);
      k_init_out<<<(int)((EF + 255) / 256), 256, 0, stream>>>(hbuf, dinv, gcnb + t * F, cur, NEDGES);
      k_msg<<<(int)(((size_t)NLG * (F / 4) + 255) / 256), 256, 0, stream>>>(hbuf, dinv, lg_src, lg_dst, cur, NLG);
      k_relu_res<<<(int)((EF + 255) / 256), 256, 0, stream>>>(base, cur, EF);
      att_pool(cur, att_Wrel, att_Wroot, att_brel, praw);
      gemm(praw, lingoutW, goutb + ((size_t)(v * NITER + t)) * GFsz, NGRAPHS, F, F,
           lingoutb, 1.f, /*tanh*/ 1, 0, 0);
      prev = cur;
    }
    k_iter_scores<<<(NGRAPHS + 255) / 256, 256, 0, stream>>>(
        goutb + (size_t)v * NITER * GFsz, lg_a, lg_abias,
        itsc + (size_t)v * NGRAPHS * NITER, NGRAPHS);
    k_combine<<<(int)((EF + 255) / 256), 256, 0, stream>>>(
        bufs + (size_t)v * NITER * EF, bufs + ((size_t)v * NITER + 1) * EF,
        bufs + ((size_t)v * NITER + 2) * EF, batch,
        itsc + (size_t)v * NGRAPHS * NITER, NEDGES);
  }
  float* out_final  = bufs;                      // view 0, slot 0 (combined in place)
  float* nout_final = bufs + (size_t)NITER * EF;

  // -------- phase C: top attention pools + contrastive loss --------
  att_pool(out_final,  top_Wrel, top_Wroot, top_brel, g1);
  att_pool(nout_final, top_Wrel, top_Wroot, top_brel, g2);

  gemm(g1, cW1, hp, NGRAPHS, F, F, cb1, 1.f, /*elu*/ 2, 0, 0);
  gemm(hp, cW2, z1, NGRAPHS, F, F, cb2, 1.f, 0, 0, 0);
  k_norm_rows<<<(NGRAPHS + 7) / 8, 256, 0, stream>>>(z1, NGRAPHS, 1e-8f);
  gemm(g2, cW1, hp, NGRAPHS, F, F, cb1, 1.f, 2, 0, 0);
  gemm(hp, cW2, z2, NGRAPHS, F, F, cb2, 1.f, 0, 0, 0);
  k_norm_rows<<<(NGRAPHS + 7) / 8, 256, 0, stream>>>(z2, NGRAPHS, 1e-8f);

  gemm(z1, z2, sim, NGRAPHS, NGRAPHS, F, nullptr, 2.0f /* 1/TAU */, 0, 0, /*transB*/ 1);
  k_lse_row<<<NGRAPHS, 256, 0, stream>>>(sim, rowt, NGRAPHS);
  { dim3 b(32, 8); k_lse_col<<<NGRAPHS / 32, b, 0, stream>>>(sim, colt, NGRAPHS); }
  k_loss<<<1, 256, 0, stream>>>(rowt, colt, out + GFsz, NGRAPHS);

  // -------- e_g = seg_sum(concat(out, noise_out) @ lin_out) --------
  gemm(out_final,  lin_out,         hbuf, NEDGES, F, F, nullptr, 1.f, 0, 0, 0);
  gemm(nout_final, lin_out + F * F, hbuf, NEDGES, F, F, nullptr, 1.f, 0, 1, 0);
  k_pool<<<NGRAPHS, 128, 0, stream>>>(hbuf, nullptr, offs, out);
}